// EBSTM_54614804136195
// MI455X (gfx1250) — compile-verified
//
#include <hip/hip_runtime.h>
#include <math.h>

typedef float v2f __attribute__((ext_vector_type(2)));
typedef float v4f __attribute__((ext_vector_type(4)));
typedef float v8f __attribute__((ext_vector_type(8)));

#define KA 64
#define CC 64
#define NPAIR 2016   // 64*63/2

// offset of row i in triu enumeration: i*63 - i*(i-1)/2 = i*(127-i)/2 (exact int)
__device__ __forceinline__ int triu_off(int i) { return (i * (127 - i)) >> 1; }

// ---------------------------------------------------------------------------
// Kernel 1: candidate generation (exact sequential slot semantics via scan)
// ---------------------------------------------------------------------------
__global__ void __launch_bounds__(256)
gen_kernel(const float* __restrict__ A_t, const float* __restrict__ feat,
           const float* __restrict__ amask, const float* __restrict__ e_dir,
           const float* __restrict__ e_none,
           int* __restrict__ rec_i, int* __restrict__ rec_j,
           float* __restrict__ rec_vij, float* __restrict__ rec_vji,
           float* __restrict__ rec_delta, float* __restrict__ out_mask) {
    const int b = blockIdx.x;
    const int t = threadIdx.x;
    const float* A  = A_t   + (size_t)b * KA * KA;
    const float* F  = feat  + (size_t)b * KA * 16;
    const float* M  = amask + (size_t)b * KA;
    const float* ED = e_dir + (size_t)b * KA * KA;
    const float* EN = e_none+ (size_t)b * KA * KA;

    __shared__ int sc[256];

    // ---- pass 1: codes for my 8 contiguous pairs ----
    const int p0 = t * 8;
    unsigned char code[8];
    signed char ia[8], ja[8];
    int nloc = 0;
    if (p0 < NPAIR) {
        float disc = sqrtf(16129.0f - 8.0f * (float)p0);
        int i = (int)((127.0f - disc) * 0.5f);
        if (i < 0) i = 0; if (i > 62) i = 62;
        while (i < 62 && triu_off(i + 1) <= p0) i++;
        while (i > 0 && triu_off(i) > p0) i--;
        int j = i + 1 + (p0 - triu_off(i));
#pragma unroll
        for (int r = 0; r < 8; ++r) {
            bool a_ij = A[i * KA + j] > 0.5f;
            bool a_ji = A[j * KA + i] > 0.5f;
            float dx = F[i * 16 + 0] - F[j * 16 + 0];
            float dy = F[i * 16 + 1] - F[j * 16 + 1];
            bool valid = (M[i] >= 0.5f) && (M[j] >= 0.5f) &&
                         (dx * dx + dy * dy <= 900.0f);
            code[r] = (unsigned char)((a_ij ? 1 : 0) | (a_ji ? 2 : 0) | (valid ? 4 : 0));
            ia[r] = (signed char)i; ja[r] = (signed char)j;
            nloc += valid ? ((a_ij && a_ji) ? 1 : 2) : 0;
            j++; if (j == KA) { i++; j = i + 1; }
        }
    }

    // ---- inclusive scan of n_per chunk sums ----
    sc[t] = nloc;
    __syncthreads();
    for (int off = 1; off < 256; off <<= 1) {
        int v = (t >= off) ? sc[t - off] : 0;
        __syncthreads();
        sc[t] += v;
        __syncthreads();
    }
    const int excl = sc[t] - nloc;
    const int total = min(1 + sc[255], CC);

    // ---- init records + candidate_mask ----
    if (t < CC) {
        int idx = b * CC + t;
        rec_i[idx] = -1; rec_j[idx] = 0;
        rec_vij[idx] = 0.0f; rec_vji[idx] = 0.0f; rec_delta[idx] = 0.0f;
        out_mask[idx] = (t < total) ? 1.0f : 0.0f;
    }
    __syncthreads();

    // ---- pass 2: walk chunk in order, emit slot records ----
    if (p0 < NPAIR) {
        int run = excl;
#pragma unroll
        for (int r = 0; r < 8; ++r) {
            bool a_ij = code[r] & 1, a_ji = (code[r] >> 1) & 1, valid = (code[r] >> 2) & 1;
            bool both = a_ij && a_ji;
            int n = valid ? (both ? 1 : 2) : 0;
            if (valid) {
                int s1 = 1 + run, s2 = 2 + run;
                bool need = (s1 < CC) || (!both && s2 < CC);
                if (need) {
                    int i = ia[r], j = ja[r];
                    float ed_ij = ED[i * KA + j];
                    float ed_ji = ED[j * KA + i];
                    float en_ij = EN[i * KA + j];
                    // valid => both agents masked-in => pair_mask true
                    float oldc = a_ij ? ed_ij : (a_ji ? ed_ji : en_ij);
                    if (s1 < CC) {
                        bool v1 = !a_ij && !a_ji;          // add edge i->j (else remove)
                        float nw = v1 ? ed_ij : en_ij;      // new A[j][i]=0 always
                        int idx = b * CC + s1;
                        rec_i[idx] = i; rec_j[idx] = j;
                        rec_vij[idx] = v1 ? 1.0f : 0.0f; rec_vji[idx] = 0.0f;
                        rec_delta[idx] = nw - oldc;
                    }
                    if (!both && s2 < CC) {
                        bool v2ij = a_ji && !a_ij;
                        bool v2ji = !a_ji;
                        float nw = v2ij ? ed_ij : (v2ji ? ed_ji : en_ij);
                        int idx = b * CC + s2;
                        rec_i[idx] = i; rec_j[idx] = j;
                        rec_vij[idx] = v2ij ? 1.0f : 0.0f;
                        rec_vji[idx] = v2ji ? 1.0f : 0.0f;
                        rec_delta[idx] = nw - oldc;
                    }
                }
            }
            run += n;
        }
    }
}

// ---------------------------------------------------------------------------
// Kernel 2: base energy E0[b] via WMMA ones-matmul reduction (1 wave / batch)
// i = q is wave-uniform, j is a per-lane constant. All loads unconditional
// (int-index addressing, saddr+voffset form); the 3-way select is an exact
// {0,1}-weighted arithmetic blend so no load is control-dependent.
// ---------------------------------------------------------------------------
__global__ void __launch_bounds__(32)
e0_kernel(const float* __restrict__ A_t, const float* __restrict__ amask,
          const float* __restrict__ e_dir, const float* __restrict__ e_none,
          float* __restrict__ E0) {
    const int b = blockIdx.x;
    const int lane = threadIdx.x;
    const float* A  = A_t   + (size_t)b * KA * KA;
    const float* M  = amask + (size_t)b * KA;
    const float* ED = e_dir + (size_t)b * KA * KA;
    const float* EN = e_none+ (size_t)b * KA * KA;

    // A-matrix f32 16x4 layout: lane<16 -> M=lane, K=0(v0)/1(v1); lane>=16 -> M=lane-16, K=2/3
    const int m  = lane & 15;
    const int kb = (lane >> 4) << 1;      // 0 or 2
    const int j0 = m * 4 + kb;            // loop-invariant per-lane columns
    const int j1 = j0 + 1;
    const int cb0 = j0 * KA;              // column base indices (per-lane const)
    const int cb1 = j1 * KA;
    const float mg0 = (M[j0] > 0.5f) ? 1.0f : 0.0f;
    const float mg1 = (M[j1] > 0.5f) ? 1.0f : 0.0f;

    v8f acc = {0.f, 0.f, 0.f, 0.f, 0.f, 0.f, 0.f, 0.f};
    v2f ones = {1.0f, 1.0f};

#pragma unroll 4
    for (int q = 0; q < 64; ++q) {
        const int rb = q * KA;                          // uniform row base
        const float rowg = (M[q] > 0.5f) ? 1.0f : 0.0f; // uniform (scalar)

        float aR0 = A[rb + j0],  aR1 = A[rb + j1];
        float aC0 = A[cb0 + q],  aC1 = A[cb1 + q];
        float eR0 = ED[rb + j0], eR1 = ED[rb + j1];
        float eC0 = ED[cb0 + q], eC1 = ED[cb1 + q];
        float nR0 = EN[rb + j0], nR1 = EN[rb + j1];

        // exact {0,1}-weight blend: no control-dependent loads
        float cR0 = (aR0 > 0.5f) ? 1.0f : 0.0f;
        float cC0 = (aC0 > 0.5f) ? 1.0f : 0.0f;
        float cR1 = (aR1 > 0.5f) ? 1.0f : 0.0f;
        float cC1 = (aC1 > 0.5f) ? 1.0f : 0.0f;
        float in0 = cC0 * eC0 + (1.0f - cC0) * nR0;
        float in1 = cC1 * eC1 + (1.0f - cC1) * nR1;
        float sel0 = cR0 * eR0 + (1.0f - cR0) * in0;
        float sel1 = cR1 * eR1 + (1.0f - cR1) * in1;

        float g0 = (q < j0) ? rowg * mg0 : 0.0f;
        float g1 = (q < j1) ? rowg * mg1 : 0.0f;

        v2f a = {sel0 * g0, sel1 * g1};
        // D[m][n] += sum_k A[m][k] : matrix-pipe reduction with ones B
        acc = __builtin_amdgcn_wmma_f32_16x16x4_f32(
            false, a, false, ones, (short)0, acc, false, false);
    }
    float s = acc[0] + acc[1] + acc[2] + acc[3] + acc[4] + acc[5] + acc[6] + acc[7];
    s += __shfl_xor(s, 16);               // rows 0-7 part + rows 8-15 part
    if (lane == 0) E0[b] = s;
}

// ---------------------------------------------------------------------------
// Kernel 3: energies + masked softmax
// ---------------------------------------------------------------------------
__global__ void __launch_bounds__(64)
fin_kernel(const float* __restrict__ E0, const float* __restrict__ rec_delta,
           const float* __restrict__ out_mask, float* __restrict__ out_energy,
           float* __restrict__ out_probs) {
    const int b = blockIdx.x, c = threadIdx.x;
    __shared__ float red[CC];
    float e = E0[b] + rec_delta[b * CC + c];
    float mk = out_mask[b * CC + c];
    float logit = (mk > 0.5f) ? -e : -1.0e9f;
    out_energy[b * CC + c] = e;

    red[c] = logit; __syncthreads();
    for (int off = 32; off > 0; off >>= 1) {
        if (c < off) red[c] = fmaxf(red[c], red[c + off]);
        __syncthreads();
    }
    float mx = red[0]; __syncthreads();
    float ex = expf(logit - mx);
    red[c] = ex; __syncthreads();
    for (int off = 32; off > 0; off >>= 1) {
        if (c < off) red[c] += red[c + off];
        __syncthreads();
    }
    out_probs[b * CC + c] = ex / red[0];
}

// ---------------------------------------------------------------------------
// Kernel 4: bulk candidate_A = A_t patched at (i,j)/(j,i)   [the 134 MB write]
// Streaming output is write-once -> nontemporal b128 stores; A_t source rows
// are reused 64x -> regular (L2-cached) loads.
// ---------------------------------------------------------------------------
__global__ void __launch_bounds__(256)
build_kernel(const float* __restrict__ A_t, const int* __restrict__ rec_i,
             const int* __restrict__ rec_j, const float* __restrict__ rec_vij,
             const float* __restrict__ rec_vji, float* __restrict__ cand) {
    const int bc = blockIdx.x;
    const int b = bc >> 6;
    const int ri = rec_i[bc], rj = rec_j[bc];
    const int p1 = (ri < 0) ? -1 : ri * KA + rj;
    const int p2 = (ri < 0) ? -1 : rj * KA + ri;
    const float vij = rec_vij[bc], vji = rec_vji[bc];

    const v4f* src = (const v4f*)(A_t + (size_t)b * KA * KA);
    v4f* dst = (v4f*)(cand + (size_t)bc * KA * KA);
    const int t = threadIdx.x;
#pragma unroll
    for (int r = 0; r < 4; ++r) {
        int vidx = t + r * 256;               // 1024 x b128 per matrix
        v4f v = src[vidx];
        int e = vidx * 4;
        v[0] = (e     == p1) ? vij : ((e     == p2) ? vji : v[0]);
        v[1] = (e + 1 == p1) ? vij : ((e + 1 == p2) ? vji : v[1]);
        v[2] = (e + 2 == p1) ? vij : ((e + 2 == p2) ? vji : v[2]);
        v[3] = (e + 3 == p1) ? vij : ((e + 3 == p2) ? vji : v[3]);
        __builtin_nontemporal_store(v, &dst[vidx]);
    }
}

// ---------------------------------------------------------------------------
extern "C" void kernel_launch(void* const* d_in, const int* in_sizes, int n_in,
                              void* d_out, int out_size, void* d_ws, size_t ws_size,
                              hipStream_t stream) {
    const float* A_t    = (const float*)d_in[0];
    const float* feat   = (const float*)d_in[1];
    const float* amask  = (const float*)d_in[2];
    const float* e_dir  = (const float*)d_in[3];
    const float* e_none = (const float*)d_in[4];
    // d_in[5] (pair_mask) intentionally unused: recomputed from agent_mask

    const int B = in_sizes[0] / (KA * KA);

    float* out = (float*)d_out;
    float* out_cand   = out;
    float* out_mask   = out + (size_t)B * CC * KA * KA;
    float* out_energy = out_mask + (size_t)B * CC;
    float* out_probs  = out_energy + (size_t)B * CC;

    // workspace partition (~164 KB for B=128)
    int*   rec_i     = (int*)d_ws;
    int*   rec_j     = rec_i + (size_t)B * CC;
    float* rec_vij   = (float*)(rec_j + (size_t)B * CC);
    float* rec_vji   = rec_vij + (size_t)B * CC;
    float* rec_delta = rec_vji + (size_t)B * CC;
    float* E0        = rec_delta + (size_t)B * CC;

    gen_kernel<<<B, 256, 0, stream>>>(A_t, feat, amask, e_dir, e_none,
                                      rec_i, rec_j, rec_vij, rec_vji,
                                      rec_delta, out_mask);
    e0_kernel<<<B, 32, 0, stream>>>(A_t, amask, e_dir, e_none, E0);
    fin_kernel<<<B, 64, 0, stream>>>(E0, rec_delta, out_mask, out_energy, out_probs);
    build_kernel<<<(B * CC), 256, 0, stream>>>(A_t, rec_i, rec_j, rec_vij,
                                               rec_vji, out_cand);
}